// GraphPoolMol_89653147337353
// MI455X (gfx1250) — compile-verified
//
#include <hip/hip_runtime.h>

// GraphPoolMol: masked neighbor max-pool.
//   B=64 batches, A=128 atoms, F=128 feats, Laplacian density ~0.1 (+identity diag).
// Strategy: 1 workgroup per batch. TDM-DMA the 64KB feature tile into LDS once
// (tensor_load_to_lds, TENSORcnt); overlap the DMA with global_prefetch_b8 of
// this wave's Laplacian rows; build per-row 128-bit neighbor masks with wave32
// ballots; sparse max-gather feature rows from LDS (ds_load_b128).

#define NB 64
#define NA 128
#define NF 128

typedef unsigned int u32x4 __attribute__((ext_vector_type(4)));
typedef int          i32x8 __attribute__((ext_vector_type(8)));
typedef int          i32x4 __attribute__((ext_vector_type(4)));

__global__ __launch_bounds__(256) void
GraphPoolMol_kernel(const float* __restrict__ nf,
                    const float* __restrict__ lap,
                    const int*   __restrict__ mol_slice,
                    float*       __restrict__ out)
{
    __shared__ __align__(128) float xs[NA * NF];   // 64 KB feature tile

    const int b    = blockIdx.x;
    const int tid  = threadIdx.x;
    const int wave = tid >> 5;
    const int lane = tid & 31;
    const int n    = mol_slice[2 * b];             // mol_slice[b, 0] = n_atoms

    const float* nf_b  = nf  + (size_t)b * NA * NF;
    const float* lap_b = lap + (size_t)b * NA * NA;

    // ---- Phase 1a: TDM load of 64KB feature tile into LDS (wave 0 issues) ----
    if (wave == 0) {
        unsigned long long ga = (unsigned long long)(uintptr_t)nf_b;
        unsigned int lds_off  = (unsigned int)(uintptr_t)(&xs[0]); // low 32b = LDS offset

        // D# group 0 (CDNA5 ISA): count=1, lds_addr, global_addr, type=2
        u32x4 g0;
        g0[0] = 1u;                                      // count=1 (valid descriptor)
        g0[1] = lds_off;                                 // lds_addr
        g0[2] = (unsigned int)(ga & 0xFFFFFFFFu);        // global_addr[31:0]
        g0[3] = (unsigned int)((ga >> 32) & 0x1FFFFFFu)  // global_addr[56:32]
              | (2u << 30);                              // type=2 ("image")

        // D# group 1 (§8.4): data_size=4B; tensor 16384x1; tile 16384x1; strides 16384
        i32x8 g1;
        g1[0] = (int)(2u << 16);        // workgroup_mask=0, data_size=2 (4B)
        g1[1] = (int)(0x4000u << 16);   // tensor_dim0[15:0]=16384 at bits[79:48]->d1[31:16]
        g1[2] = (int)(1u << 16);        // tensor_dim0 hi=0; tensor_dim1[15:0]=1
        g1[3] = (int)(0x4000u << 16);   // tensor_dim1 hi=0; tile_dim0=16384
        g1[4] = 1;                      // tile_dim1=1, tile_dim2=0
        g1[5] = 16384;                  // tensor_dim0_stride[31:0]
        g1[6] = (int)(0x4000u << 16);   // stride0 hi=0; tensor_dim1_stride[15:0]=16384
        g1[7] = 0;                      // tensor_dim1_stride hi=0

        i32x4 g2 = {0, 0, 0, 0};        // groups 2/3 unused (<=2D tensor)
        i32x4 g3 = {0, 0, 0, 0};
        i32x8 g4 = {0, 0, 0, 0, 0, 0, 0, 0};  // extra operand of 6-arg builtin (unused)

        __builtin_amdgcn_tensor_load_to_lds(g0, g1, g2, g3, g4, 0);
    }

    // ---- Phase 1b: overlap the DMA — prefetch this wave's Laplacian rows.
    // Each wave owns rows {wave, wave+8, ...}; lanes cover the 512B row at
    // cacheline granularity (global_prefetch_b8, no counters, no VGPR return).
    for (int r = wave; r < NA; r += 8) {
        __builtin_prefetch(lap_b + (size_t)r * NA + lane * 4, 0, 3);
    }

    if (wave == 0) {
        __builtin_amdgcn_s_wait_tensorcnt(0);  // LDS tile complete before barrier
    }
    __syncthreads();   // LDS tile visible to all 8 waves

    // ---- Phase 2: per-row sparse max-pool. Each wave owns 16 rows. ----
    const int f0 = lane << 2;  // 4 contiguous feats per lane -> b128 LDS/global ops
    for (int r = wave; r < NA; r += 8) {
        float4 acc;
        if (r < n) {
            // Build 128-bit neighbor mask for row r (wave-uniform words).
            const float* lrow = lap_b + (size_t)r * NA;
            unsigned mm[4];
#pragma unroll
            for (int g = 0; g < 4; ++g) {
                int col = (g << 5) + lane;
                float Lv = lrow[col];
                mm[g] = __builtin_amdgcn_ballot_w32((Lv != 0.0f) && (col < n));
            }
            if ((mm[0] | mm[1] | mm[2] | mm[3]) == 0u) {
                // no neighbors -> passthrough own features
                acc = *(const float4*)(xs + r * NF + f0);
            } else {
                acc.x = acc.y = acc.z = acc.w = -__builtin_inff();
#pragma unroll
                for (int g = 0; g < 4; ++g) {
                    unsigned bits = mm[g];           // uniform -> scalar loop control
                    while (bits) {
                        int j = (g << 5) + __builtin_ctz(bits);
                        bits &= bits - 1u;
                        float4 v = *(const float4*)(xs + j * NF + f0);
                        acc.x = fmaxf(acc.x, v.x);
                        acc.y = fmaxf(acc.y, v.y);
                        acc.z = fmaxf(acc.z, v.z);
                        acc.w = fmaxf(acc.w, v.w);
                    }
                }
            }
        } else {
            acc.x = acc.y = acc.z = acc.w = 0.0f;    // invalid rows -> 0
        }
        *(float4*)(out + ((size_t)b * NA + r) * NF + f0) = acc;
    }
}

extern "C" void kernel_launch(void* const* d_in, const int* in_sizes, int n_in,
                              void* d_out, int out_size, void* d_ws, size_t ws_size,
                              hipStream_t stream) {
    (void)in_sizes; (void)n_in; (void)out_size; (void)d_ws; (void)ws_size;
    const float* nf  = (const float*)d_in[0];   // node_features (64,128,128) f32
    const float* lap = (const float*)d_in[1];   // laplacian     (64,128,128) f32
    const int*   mol = (const int*)d_in[2];     // mol_slice     (64,2) i32
    float* out = (float*)d_out;                 // (64,128,128) f32
    GraphPoolMol_kernel<<<dim3(NB), dim3(256), 0, stream>>>(nf, lap, mol, out);
}